// Encoder_48619029790897
// MI455X (gfx1250) — compile-verified
//
#include <hip/hip_runtime.h>

// Problem constants (from the reference)
#define N_NODES 100000
#define N_EDGES 1600000
#define IN_DIM  256
#define HID     128
#define OUTD    64

typedef __attribute__((ext_vector_type(2))) float v2f;
typedef __attribute__((ext_vector_type(8))) float v8f;

// ---------------------------------------------------------------------------
// Degree / normalization kernels
// ---------------------------------------------------------------------------
__global__ void k_init_deg(float* __restrict__ deg) {
  int i = blockIdx.x * blockDim.x + threadIdx.x;
  if (i < N_NODES) deg[i] = 1.0f;  // self-loop counted
}

__global__ void k_count_deg(const int* __restrict__ ei, float* __restrict__ deg) {
  int e = blockIdx.x * blockDim.x + threadIdx.x;
  if (e < N_EDGES) {
    int d = ei[N_EDGES + e];  // dst row of edge_index
    unsafeAtomicAdd(&deg[d], 1.0f);  // global_atomic_add_f32 (L2 atomic unit)
  }
}

__global__ void k_rsqrt_inplace(float* __restrict__ deg) {
  int i = blockIdx.x * blockDim.x + threadIdx.x;
  if (i < N_NODES) deg[i] = rsqrtf(deg[i]);  // now holds dinv
}

// ---------------------------------------------------------------------------
// fp32 WMMA GEMM: C[M, NT*16] = (RELU? relu(A) : A)[M, K] @ W[K, NT*16]
// One wave -> 16 output rows x full width (NT accumulators of v8f).
//
// W is staged through LDS in 64-row chunks using a fragment-ready interleaved
// layout:  lds[(k/2)*(2*NC) + 2*n + (k&1)] = W[k][n]
// so each per-lane B fragment {W[k][n], W[k+1][n]} is one aligned ds_load_b64.
//
// V_WMMA_F32_16X16X4_F32 per-lane layout (ISA 7.12.2):
//   A 16x4 : m = lane&15, VGPR0/1 = K = 2*(lane>>4) + {0,1}
//   B 4x16 : n = lane&15, VGPR0/1 = K = 2*(lane>>4) + {0,1}
//   C 16x16: VGPR r holds row r + 8*(lane>>4), col = lane&15
// ---------------------------------------------------------------------------
template <int K, int NT, bool RELU>
__global__ __launch_bounds__(256) void k_gemm_wmma(const float* __restrict__ A,
                                                   const float* __restrict__ W,
                                                   float* __restrict__ C, int M) {
  constexpr int NC = NT * 16;   // output columns
  constexpr int KC = 64;        // K-chunk staged in LDS
  __shared__ float lds[KC * NC];

  const int wave = threadIdx.x >> 5;
  const int lane = threadIdx.x & 31;
  const int mtile = blockIdx.x * 8 + wave;
  const long m0 = (long)mtile * 16;
  const bool active = (m0 < M);

  const int m  = lane & 15;   // A row within tile
  const int kh = lane >> 4;   // K half select
  const int n  = lane & 15;   // B/C column within tile

  // Clamp so inactive waves still issue safe loads (they must reach barriers).
  const long arow = active ? (m0 + m) : (long)(M - 1);
  const float* __restrict__ Aptr = A + arow * K;

  v8f acc[NT] = {};

  for (int kc = 0; kc < K; kc += KC) {
    if (kc != 0) __syncthreads();
    // Cooperative stage of W[kc..kc+KC) into interleaved-pair LDS layout.
    // Each entry covers 2 rows x 4 cols: two float4 global loads (coalesced),
    // two aligned ds_store_b128 of the interleave.
    {
      constexpr int ENT = KC * NC / 8;  // (KC/2 pairs) * (NC/4 col quads)
#pragma unroll
      for (int idx = threadIdx.x; idx < ENT; idx += 256) {
        int kp = idx / (NC / 4);
        int nq = (idx % (NC / 4)) * 4;
        const float4 a = *(const float4*)(W + (long)(kc + 2 * kp) * NC + nq);
        const float4 b = *(const float4*)(W + (long)(kc + 2 * kp + 1) * NC + nq);
        float4* __restrict__ dst = (float4*)&lds[kp * 2 * NC + 2 * nq];
        dst[0] = make_float4(a.x, b.x, a.y, b.y);
        dst[1] = make_float4(a.z, b.z, a.w, b.w);
      }
    }
    __syncthreads();

#pragma unroll 4
    for (int k0 = kc; k0 < kc + KC; k0 += 4) {
      const int ka = k0 + 2 * kh;       // global K of this lane's A/B pair
      const int kp = (ka - kc) >> 1;    // LDS-local pair index
      v2f av = *(const v2f*)(Aptr + ka);
      if (RELU) {
        av.x = fmaxf(av.x, 0.0f);
        av.y = fmaxf(av.y, 0.0f);
      }
      const v2f* __restrict__ brow = (const v2f*)&lds[kp * 2 * NC + 2 * n];
#pragma unroll
      for (int j = 0; j < NT; ++j) {
        v2f bv = brow[j * 16];  // single ds_load_b64, offset j*128 bytes
        acc[j] = __builtin_amdgcn_wmma_f32_16x16x4_f32(
            /*neg_a=*/false, av, /*neg_b=*/false, bv,
            /*c_mod=*/(short)0, acc[j], /*reuse_a=*/false, /*reuse_b=*/false);
      }
    }
  }

  if (!active) return;
#pragma unroll
  for (int j = 0; j < NT; ++j) {
#pragma unroll
    for (int r = 0; r < 8; ++r) {
      long row = m0 + r + 8 * kh;
      if (row < M) C[row * NC + j * 16 + n] = acc[j][r];
    }
  }
}

// ---------------------------------------------------------------------------
// agg[i,f] = h[i,f] * dinv[i]^2 + b[f]   (self-loop contribution + bias)
// ---------------------------------------------------------------------------
template <int F>
__global__ void k_self_init(const float* __restrict__ h,
                            const float* __restrict__ dinv,
                            const float* __restrict__ b,
                            float* __restrict__ agg) {
  long t = (long)blockIdx.x * blockDim.x + threadIdx.x;
  if (t >= (long)N_NODES * F) return;
  long i = t / F;
  int f = (int)(t % F);
  float di = dinv[i];
  agg[t] = h[t] * (di * di) + b[f];
}

// ---------------------------------------------------------------------------
// Edge aggregation: per (edge, float4-chunk) thread.
//   agg[dst, :] += h[src, :] * dinv[src]*dinv[dst]
// Dominant traffic; agg fits in 192 MB L2 so atomics stay on-chip.
// ---------------------------------------------------------------------------
template <int F>
__global__ void k_edge_agg(const int* __restrict__ ei,
                           const float* __restrict__ dinv,
                           const float* __restrict__ h,
                           float* __restrict__ agg) {
  constexpr int CH = F / 4;  // float4 chunks per edge
  long t = (long)blockIdx.x * blockDim.x + threadIdx.x;
  long e = t / CH;
  int c = (int)(t % CH);
  if (e >= N_EDGES) return;
  int s = ei[e];
  int d = ei[N_EDGES + e];
  float norm = dinv[s] * dinv[d];
  float4 v = *(const float4*)(h + (long)s * F + c * 4);
  float* __restrict__ out = agg + (long)d * F + c * 4;
  unsafeAtomicAdd(out + 0, v.x * norm);
  unsafeAtomicAdd(out + 1, v.y * norm);
  unsafeAtomicAdd(out + 2, v.z * norm);
  unsafeAtomicAdd(out + 3, v.w * norm);
}

// ---------------------------------------------------------------------------
// Launch
// ---------------------------------------------------------------------------
extern "C" void kernel_launch(void* const* d_in, const int* in_sizes, int n_in,
                              void* d_out, int out_size, void* d_ws, size_t ws_size,
                              hipStream_t stream) {
  const float* x  = (const float*)d_in[0];
  const int*   ei = (const int*)d_in[1];  // edge_index [2, E] flat
  const float* W1 = (const float*)d_in[2];
  const float* b1 = (const float*)d_in[3];
  const float* W2 = (const float*)d_in[4];
  const float* b2 = (const float*)d_in[5];
  float* out = (float*)d_out;

  float* ws   = (float*)d_ws;
  float* dinv = ws;                                   // N
  float* h1   = dinv + N_NODES;                       // N*HID
  float* agg1 = h1 + (size_t)N_NODES * HID;           // N*HID
  float* h3   = agg1 + (size_t)N_NODES * HID;         // N*OUTD

  const int T = 256;

  // degrees -> dinv
  k_init_deg<<<(N_NODES + T - 1) / T, T, 0, stream>>>(dinv);
  k_count_deg<<<(N_EDGES + T - 1) / T, T, 0, stream>>>(ei, dinv);
  k_rsqrt_inplace<<<(N_NODES + T - 1) / T, T, 0, stream>>>(dinv);

  // layer 1: h1 = x @ W1 (fp32 WMMA)
  const int mtiles = (N_NODES + 15) / 16;             // 6250
  const int gblocks = (mtiles + 7) / 8;               // 8 waves/block
  k_gemm_wmma<IN_DIM, HID / 16, false>
      <<<gblocks, T, 0, stream>>>(x, W1, h1, N_NODES);

  // agg1 = h1*dinv^2 + b1, then += scatter of h1[src]*norm
  {
    long tot = (long)N_NODES * HID;
    k_self_init<HID><<<(unsigned)((tot + T - 1) / T), T, 0, stream>>>(h1, dinv, b1, agg1);
    long et = (long)N_EDGES * (HID / 4);
    k_edge_agg<HID><<<(unsigned)((et + T - 1) / T), T, 0, stream>>>(ei, dinv, h1, agg1);
  }

  // layer 2: h3 = relu(agg1) @ W2 (ReLU fused into A-fragment load)
  k_gemm_wmma<HID, OUTD / 16, true>
      <<<gblocks, T, 0, stream>>>(agg1, W2, h3, N_NODES);

  // out = h3*dinv^2 + b2, then += scatter of h3[src]*norm
  {
    long tot = (long)N_NODES * OUTD;
    k_self_init<OUTD><<<(unsigned)((tot + T - 1) / T), T, 0, stream>>>(h3, dinv, b2, out);
    long et = (long)N_EDGES * (OUTD / 4);
    k_edge_agg<OUTD><<<(unsigned)((et + T - 1) / T), T, 0, stream>>>(ei, dinv, h3, out);
  }
}